// Soft_cldice_80942953660574
// MI455X (gfx1250) — compile-verified
//
#include <hip/hip_runtime.h>
#include <hip/hip_bf16.h>
#include <stdint.h>

// Problem geometry: (N=32, C=1, H=1024, W=1024) fp32
#define IMH 1024
#define IMW 1024
#define NSLICE 32
#define NTOT (NSLICE * IMH * IMW)

// 2-step fused tiling: 64x64 output tile.
// Only the erode chain propagates, so 2 fused steps need halo 3:
//   a_t = e_k     over tile+3  (70x70), OOB = +INF (min-pool padding)
//   b_t = e_{k+1} over tile+2  (68x68), OOB = -1 sentinel
//   c_t = e_{k+2} over tile+1  (66x66), OOB = -1 sentinel
#define TW 64
#define TH 64
#define AW (TW + 6)   // 70
#define AH (TH + 6)
#define BW (TW + 4)   // 68
#define BH (TH + 4)
#define CW (TW + 2)   // 66
#define CH (TH + 2)

// Reduction: elements per block / number of partial blocks
#define RB 8192
#define NBLK (NTOT / RB)   // 4096

#define FINF __builtin_inff()

typedef __attribute__((ext_vector_type(2))) float v2f;
typedef __attribute__((ext_vector_type(8))) float v8f;

// ---------------- CDNA5 async global->LDS copy (ASYNCcnt path) ----------------
__device__ __forceinline__ void async_ld_b32(const float* g, float* l) {
#if defined(__has_builtin) && __has_builtin(__builtin_amdgcn_global_load_async_to_lds_b32)
  __builtin_amdgcn_global_load_async_to_lds_b32(
      (__attribute__((address_space(1))) int*)(g),
      (__attribute__((address_space(3))) int*)(l),
      0, 0);
#else
  *l = *g;  // fallback: global_load + ds_store
#endif
}

__device__ __forceinline__ void wait_async0() {
#if defined(__has_builtin) && __has_builtin(__builtin_amdgcn_s_wait_asynccnt)
  __builtin_amdgcn_s_wait_asynccnt(0);
#else
  asm volatile("s_wait_asynccnt 0" ::: "memory");
#endif
}

// ---------------- wave32 sum via V_WMMA_F32_16X16X4_F32 ----------------
// A holds the 32 lane values (second A reg = 0), B = ones. Every column of
// D = A x ones equals the row-sum vector of A, and the sum of one full column
// (rows 0..7 in lane L's 8 D regs, rows 8..15 in lane L+16's) equals the sum of
// ALL elements of A -- invariant to the A-layout permutation.
__device__ __forceinline__ float wave32_sum(float x) {
#if defined(__has_builtin) && __has_builtin(__builtin_amdgcn_wmma_f32_16x16x4_f32)
  v2f a; a[0] = x;    a[1] = 0.0f;
  v2f b; b[0] = 1.0f; b[1] = 1.0f;
  v8f c = {0.f, 0.f, 0.f, 0.f, 0.f, 0.f, 0.f, 0.f};
  v8f d = __builtin_amdgcn_wmma_f32_16x16x4_f32(false, a, false, b, (short)0, c,
                                                false, false);
  float cs = d[0] + d[1] + d[2] + d[3] + d[4] + d[5] + d[6] + d[7];
  return __shfl(cs, 0, 32) + __shfl(cs, 16, 32);
#else
  for (int o = 16; o; o >>= 1) x += __shfl_xor(x, o, 32);
  return x;
#endif
}

// map a -1 OOB sentinel to +INF for min-pool consumption
__device__ __forceinline__ float min_tap(float t) { return t < 0.0f ? FINF : t; }

// ---------------- fused 2-step morphology kernel ----------------
// Performs chain steps k=2p and k=2p+1 in one pass:
//   e1 = erode(e_k); e2 = erode(e1)
//   d0 = relu(e_k - dilate3x3(e1)); skel = init ? d0 : skel + relu(d0 - skel*d0)
//   d1 = relu(e1  - dilate3x3(e2)); skel = skel + relu(d1 - skel*d1)
//   (write_next) enext = e2
__global__ __launch_bounds__(256) void step2_kernel(
    const float* __restrict__ src, float* __restrict__ enext,
    float* __restrict__ skel, int init, int write_next) {
  __shared__ float a_t[AH * AW];
  __shared__ float b_t[BH * BW];
  __shared__ float c_t[CH * CW];

  const int tid = threadIdx.x;
  const int x0 = blockIdx.x * TW;
  const int y0 = blockIdx.y * TH;
  const int z  = blockIdx.z;
  const float* sp = src + (size_t)z * IMH * IMW;

  // Stage e_k tile (+halo 3) via async global->LDS; OOB = +INF.
  for (int li = tid; li < AH * AW; li += 256) {
    int ly = li / AW, lx = li - ly * AW;
    int gy = y0 - 3 + ly, gx = x0 - 3 + lx;
    if ((unsigned)gy < IMH && (unsigned)gx < IMW) {
      async_ld_b32(sp + (size_t)gy * IMW + gx, &a_t[li]);
    } else {
      a_t[li] = FINF;
    }
  }
  wait_async0();
  __syncthreads();

  // e_{k+1} over 68x68 (tile+2). a_t OOB already +INF, so plain 5-tap min.
  for (int li = tid; li < BH * BW; li += 256) {
    int by = li / BW, bx = li - by * BW;
    int gy = y0 - 2 + by, gx = x0 - 2 + bx;
    float v;
    if ((unsigned)gy < IMH && (unsigned)gx < IMW) {
      const float* p = &a_t[(by + 1) * AW + (bx + 1)];
      float c = p[0];
      float p1 = fminf(fminf(p[-AW], p[AW]), c);
      float p2 = fminf(fminf(p[-1], p[1]), c);
      v = fminf(p1, p2);
    } else {
      v = -1.0f;  // sentinel: ignored by max (values >= 0), remapped for min
    }
    b_t[li] = v;
  }
  __syncthreads();

  // e_{k+2} over 66x66 (tile+1). b_t taps remapped (-1 -> +INF) for the min.
  for (int li = tid; li < CH * CW; li += 256) {
    int cy = li / CW, cx = li - cy * CW;
    int gy = y0 - 1 + cy, gx = x0 - 1 + cx;
    float v;
    if ((unsigned)gy < IMH && (unsigned)gx < IMW) {
      const float* p = &b_t[(cy + 1) * BW + (cx + 1)];
      float c = min_tap(p[0]);
      float p1 = fminf(fminf(min_tap(p[-BW]), min_tap(p[BW])), c);
      float p2 = fminf(fminf(min_tap(p[-1]), min_tap(p[1])), c);
      v = fminf(p1, p2);
    } else {
      v = -1.0f;
    }
    c_t[li] = v;
  }
  __syncthreads();

  // interior: two open/delta/skel updates, write e_{k+2}
  for (int li = tid; li < TW * TH; li += 256) {
    int py = li / TW, px = li - py * TW;

    // open of e_{k+1} (3x3 max on b_t; -1 sentinels auto-ignored)
    const float* b = &b_t[(py + 2) * BW + (px + 2)];
    float m0 = fmaxf(fmaxf(b[-BW - 1], b[-BW]), b[-BW + 1]);
    float m1 = fmaxf(fmaxf(b[-1], b[0]), b[1]);
    float m2 = fmaxf(fmaxf(b[BW - 1], b[BW]), b[BW + 1]);
    float open1 = fmaxf(fmaxf(m0, m1), m2);
    float d0 = fmaxf(a_t[(py + 3) * AW + (px + 3)] - open1, 0.0f);

    // open of e_{k+2} (3x3 max on c_t)
    const float* c = &c_t[(py + 1) * CW + (px + 1)];
    float n0 = fmaxf(fmaxf(c[-CW - 1], c[-CW]), c[-CW + 1]);
    float n1 = fmaxf(fmaxf(c[-1], c[0]), c[1]);
    float n2 = fmaxf(fmaxf(c[CW - 1], c[CW]), c[CW + 1]);
    float open2 = fmaxf(fmaxf(n0, n1), n2);
    float d1 = fmaxf(b[0] - open2, 0.0f);

    size_t gidx = (size_t)z * IMH * IMW + (size_t)(y0 + py) * IMW + (x0 + px);
    float s;
    if (init) {
      s = d0;
    } else {
      float s0 = skel[gidx];
      s = s0 + fmaxf(d0 - s0 * d0, 0.0f);
    }
    s = s + fmaxf(d1 - s * d1, 0.0f);
    skel[gidx] = s;
    if (write_next) enext[gidx] = c[0];
  }
}

// ---------------- stage-1 reduction: per-block partials ----------------
__global__ __launch_bounds__(256) void reduce_kernel(
    const float* __restrict__ skel, const float* __restrict__ other,
    float* __restrict__ part0, float* __restrict__ part1) {
  __shared__ float sh0[256];
  __shared__ float sh1[256];
  size_t base = (size_t)blockIdx.x * RB;
  float a0 = 0.f, a1 = 0.f;
  for (int i = threadIdx.x; i < RB; i += 256) {
    float s = skel[base + i];
    float o = other[base + i];
    a0 += s * o;
    a1 += s;
  }
  sh0[threadIdx.x] = a0;
  sh1[threadIdx.x] = a1;
  __syncthreads();
  for (int off = 128; off > 0; off >>= 1) {
    if (threadIdx.x < off) {
      sh0[threadIdx.x] += sh0[threadIdx.x + off];
      sh1[threadIdx.x] += sh1[threadIdx.x + off];
    }
    __syncthreads();
  }
  if (threadIdx.x == 0) {
    part0[blockIdx.x] = sh0[0];
    part1[blockIdx.x] = sh1[0];
  }
}

// ---------------- stage-2: bank sums for skel_pred ----------------
__global__ void final_partial_kernel(const float* __restrict__ part0,
                                     const float* __restrict__ part1,
                                     float* __restrict__ accum) {
  int lane = threadIdx.x;
  float a0 = 0.f, a1 = 0.f;
  for (int i = lane; i < NBLK; i += 32) { a0 += part0[i]; a1 += part1[i]; }
  float t0 = wave32_sum(a0);
  float t1 = wave32_sum(a1);
  if (lane == 0) { accum[0] = t0; accum[1] = t1; }
}

// ---------------- stage-2: skel_true sums + cl_dice scalar ----------------
__global__ void final_kernel(const float* __restrict__ part0,
                             const float* __restrict__ part1,
                             const float* __restrict__ accum,
                             float* __restrict__ out) {
  int lane = threadIdx.x;
  float a0 = 0.f, a1 = 0.f;
  for (int i = lane; i < NBLK; i += 32) { a0 += part0[i]; a1 += part1[i]; }
  float sum_tw = wave32_sum(a0);  // sum(skel_true * inputs)
  float sum_t  = wave32_sum(a1);  // sum(skel_true)
  if (lane == 0) {
    float sum_pw = accum[0];      // sum(skel_pred * target)
    float sum_p  = accum[1];      // sum(skel_pred)
    const float SM = 1e-5f;
    float tprec = (sum_pw + SM) / (sum_p + SM);
    float tsens = (sum_tw + SM) / (sum_t + SM);
    out[0] = 1.0f - 2.0f * (tprec * tsens) / (tprec + tsens);
  }
}

extern "C" void kernel_launch(void* const* d_in, const int* in_sizes, int n_in,
                              void* d_out, int out_size, void* d_ws, size_t ws_size,
                              hipStream_t stream) {
  (void)in_sizes; (void)n_in; (void)out_size; (void)ws_size;
  const float* target = (const float*)d_in[0];
  const float* inputs = (const float*)d_in[1];
  float* out = (float*)d_out;

  char* ws = (char*)d_ws;
  const size_t img_bytes = (size_t)NTOT * sizeof(float);
  float* Ea    = (float*)(ws);
  float* Eb    = (float*)(ws + img_bytes);
  float* S     = (float*)(ws + 2 * img_bytes);
  float* part0 = (float*)(ws + 3 * img_bytes);
  float* part1 = part0 + NBLK;
  float* accum = part1 + NBLK;

  dim3 grid(IMW / TW, IMH / TH, NSLICE);
  dim3 blk(256);

  // ---- chain on `inputs` -> skel_pred in S (13 fused double-steps) ----
  {
    const float* cur = inputs;
    float* bufs[2] = {Ea, Eb};
    for (int p = 0; p < 13; ++p) {
      float* nxt = bufs[p & 1];
      step2_kernel<<<grid, blk, 0, stream>>>(cur, nxt, S, (p == 0) ? 1 : 0,
                                             (p < 12) ? 1 : 0);
      cur = nxt;
    }
  }
  reduce_kernel<<<NBLK, blk, 0, stream>>>(S, target, part0, part1);
  final_partial_kernel<<<1, 32, 0, stream>>>(part0, part1, accum);

  // ---- chain on `target` -> skel_true in S (S reused) ----
  {
    const float* cur = target;
    float* bufs[2] = {Ea, Eb};
    for (int p = 0; p < 13; ++p) {
      float* nxt = bufs[p & 1];
      step2_kernel<<<grid, blk, 0, stream>>>(cur, nxt, S, (p == 0) ? 1 : 0,
                                             (p < 12) ? 1 : 0);
      cur = nxt;
    }
  }
  reduce_kernel<<<NBLK, blk, 0, stream>>>(S, inputs, part0, part1);
  final_kernel<<<1, 32, 0, stream>>>(part0, part1, accum, out);
}